// SelectiveSSM_28698971472188
// MI455X (gfx1250) — compile-verified
//
#include <hip/hip_runtime.h>
#include <hip/hip_bf16.h>
#include <math.h>

#define D_MODEL 1024
#define D_STATE 16
#define D_CONV  4
#define D_INNER 2048
#define BATCH   2
#define SEQLEN  2048
#define NROWS   (BATCH * SEQLEN)   // 4096 rows for the flattened GEMMs

typedef float v2f __attribute__((ext_vector_type(2)));
typedef float v8f __attribute__((ext_vector_type(8)));
typedef unsigned int u32x4 __attribute__((ext_vector_type(4)));
typedef int i32x4 __attribute__((ext_vector_type(4)));
typedef int i32x8 __attribute__((ext_vector_type(8)));

__device__ __forceinline__ float sigmoid_f(float v) {
    return 1.0f / (1.0f + __expf(-v));
}

// ---------------------------------------------------------------------------
// TDM: issue a 2D tile load Global -> LDS via tensor descriptor (D#).
// Tile is tile_h rows x tile_w elements (fp32), row stride = row_stride elems.
// LDS receives the tile packed [tile_h][tile_w]. Wave-level op; EXEC ignored.
// Descriptor layout per CDNA5 ISA ch.10.8 (group0 128b, group1 256b).
// Toolchain uses the 6-arg builtin: (g0, g1, g2, g3, g_extra, cpol).
// ---------------------------------------------------------------------------
__device__ __forceinline__ void tdm_load_tile_2d(unsigned lds_off,
                                                 const float* gptr,
                                                 unsigned tile_w, unsigned tile_h,
                                                 unsigned long long row_stride_elems)
{
    unsigned long long ga = (unsigned long long)(size_t)gptr;
    u32x4 g0;
    g0.x = 1u;                                          // count=1, is_restore=0, gather off
    g0.y = lds_off;                                     // lds_addr (bytes)
    g0.z = (unsigned)(ga & 0xFFFFFFFFu);                // global_addr[31:0]
    g0.w = (unsigned)((ga >> 32) & 0x01FFFFFFu)         // global_addr[56:32]
         | (2u << 30);                                  // type = 2 ("image")
    i32x8 g1;
    g1[0] = (int)(2u << 16);                            // data_size = 2 -> 4 bytes
    g1[1] = (int)((tile_w & 0xFFFFu) << 16);            // tensor_dim0[15:0] (== tile_w)
    g1[2] = (int)((tile_w >> 16) | ((tile_h & 0xFFFFu) << 16)); // dim0 hi | tensor_dim1 lo
    g1[3] = (int)((tile_h >> 16) | ((tile_w & 0xFFFFu) << 16)); // dim1 hi | tile_dim0
    g1[4] = (int)(tile_h & 0xFFFFu);                    // tile_dim1 (tile_dim2 = 0)
    g1[5] = (int)(row_stride_elems & 0xFFFFFFFFull);    // tensor_dim0_stride[31:0]
    g1[6] = (int)((row_stride_elems >> 32) & 0xFFFFull);// stride hi | stride1 lo (0)
    g1[7] = 0;
    i32x4 z4 = (i32x4)0;
    i32x8 z8 = (i32x8)0;
    __builtin_amdgcn_tensor_load_to_lds(g0, g1, z4, z4, z8, 0);
}

// ---------------------------------------------------------------------------
// GEMM: C[M,N] = A[M,K] @ B[K,N] + bias[N]   (fp32, V_WMMA_F32_16X16X4_F32)
// Block = 256 threads (8 waves), block tile 128x128, wave tile 64x32.
// Double-buffered LDS tiles staged by the Tensor Data Mover (wave 0 issues,
// TENSORcnt + workgroup barrier synchronize), overlapping DMA with WMMA.
// ---------------------------------------------------------------------------
template <int BM, int BN, int KC>
__global__ __launch_bounds__(256)
void gemm_f32_wmma(const float* __restrict__ A, const float* __restrict__ B,
                   const float* __restrict__ bias, float* __restrict__ C,
                   int M, int N, int K)
{
    __shared__ float sA[2][BM][KC];   // A tile: [m][k]
    __shared__ float sB[2][KC][BN];   // B tile: [k][n]

    const int tid  = threadIdx.x;
    const int wave = tid >> 5;
    const int lane = tid & 31;
    const int l16  = lane & 15;
    const int hi   = lane >> 4;        // which half of the wave
    const int wm   = wave >> 2;        // 0..1  (64-row stripes)
    const int wn   = wave & 3;         // 0..3  (32-col stripes)
    const int row0 = blockIdx.y * BM;
    const int col0 = blockIdx.x * BN;
    const bool tdmWave = (wave == 0);

    v8f acc[4][2];
#pragma unroll
    for (int i = 0; i < 4; ++i)
#pragma unroll
        for (int j = 0; j < 2; ++j)
            acc[i][j] = (v8f)(0.0f);

    const int nstages = K / KC;
    if (tdmWave) {
        tdm_load_tile_2d((unsigned)(size_t)&sA[0][0][0],
                         A + (size_t)row0 * K, KC, BM, (unsigned long long)K);
        tdm_load_tile_2d((unsigned)(size_t)&sB[0][0][0],
                         B + col0, BN, KC, (unsigned long long)N);
    }

    for (int s = 0; s < nstages; ++s) {
        const int cur = s & 1;
        if (tdmWave) __builtin_amdgcn_s_wait_tensorcnt(0);  // buf[cur] DMA done
        __syncthreads();   // all waves: buf[cur] ready; buf[cur^1] no longer read

        if (tdmWave && (s + 1) < nstages) {                 // prefetch next stage
            const int k0n = (s + 1) * KC;
            tdm_load_tile_2d((unsigned)(size_t)&sA[cur ^ 1][0][0],
                             A + (size_t)row0 * K + k0n, KC, BM, (unsigned long long)K);
            tdm_load_tile_2d((unsigned)(size_t)&sB[cur ^ 1][0][0],
                             B + (size_t)k0n * N + col0, BN, KC, (unsigned long long)N);
        }

#pragma unroll
        for (int ks = 0; ks < KC; ks += 4) {
            // A fragment (16x4 f32): lanes 0-15 hold K={ks,ks+1}, lanes 16-31 K={ks+2,ks+3}
            v2f afrag[4];
#pragma unroll
            for (int i = 0; i < 4; ++i) {
                const float* p = &sA[cur][wm * 64 + i * 16 + l16][ks + 2 * hi];
                afrag[i].x = p[0];
                afrag[i].y = p[1];
            }
            // B fragment (4x16 f32): same K split, N striped across lanes
            v2f bfrag[2];
#pragma unroll
            for (int j = 0; j < 2; ++j) {
                bfrag[j].x = sB[cur][ks + 2 * hi + 0][wn * 32 + j * 16 + l16];
                bfrag[j].y = sB[cur][ks + 2 * hi + 1][wn * 32 + j * 16 + l16];
            }
#pragma unroll
            for (int i = 0; i < 4; ++i)
#pragma unroll
                for (int j = 0; j < 2; ++j)
                    acc[i][j] = __builtin_amdgcn_wmma_f32_16x16x4_f32(
                        false, afrag[i], false, bfrag[j],
                        (short)0, acc[i][j], false, false);
        }
        __syncthreads();   // done reading buf[cur] before its DMA at stage s+2
    }

    // Epilogue: C/D layout — VGPR e: lanes 0-15 -> M=e, lanes 16-31 -> M=e+8; N = lane&15
#pragma unroll
    for (int i = 0; i < 4; ++i) {
#pragma unroll
        for (int j = 0; j < 2; ++j) {
            int ccol = col0 + wn * 32 + j * 16 + l16;
            float bv = bias[ccol];
#pragma unroll
            for (int e = 0; e < 8; ++e) {
                int r = row0 + wm * 64 + i * 16 + e + 8 * hi;
                C[(size_t)r * N + ccol] = acc[i][j][e] + bv;
            }
        }
    }
}

// ---------------------------------------------------------------------------
// Depthwise causal conv (width 4) + SiLU.  x_ssm = xz[:, :, 0:D_INNER]
// ---------------------------------------------------------------------------
__global__ __launch_bounds__(256)
void conv_silu_kernel(const float* __restrict__ xz, const float* __restrict__ conv_w,
                      const float* __restrict__ conv_b, float* __restrict__ xc)
{
    size_t idx = (size_t)blockIdx.x * blockDim.x + threadIdx.x;
    if (idx >= (size_t)NROWS * D_INNER) return;
    int d   = (int)(idx % D_INNER);
    int row = (int)(idx / D_INNER);          // row = b*SEQLEN + t
    int t   = row % SEQLEN;
    int b   = row / SEQLEN;

    float acc = conv_b[d];
#pragma unroll
    for (int k = 0; k < D_CONV; ++k) {
        int tt = t - (D_CONV - 1) + k;
        if (tt >= 0) {
            float xv = xz[((size_t)(b * SEQLEN + tt)) * (2 * D_INNER) + d];
            acc = fmaf(xv, conv_w[d * D_CONV + k], acc);
        }
    }
    xc[idx] = acc * sigmoid_f(acc);          // SiLU
}

// ---------------------------------------------------------------------------
// proj = xc @ W_xproj  (N = 33 = 16 B, 16 C, 1 delta).  delta stored softplus'ed.
// One block per row; row staged in LDS.
// ---------------------------------------------------------------------------
__global__ __launch_bounds__(64)
void xproj_kernel(const float* __restrict__ xc, const float* __restrict__ W_xproj,
                  float* __restrict__ proj)
{
    __shared__ float srow[D_INNER];
    int r = blockIdx.x;
    for (int i = threadIdx.x; i < D_INNER; i += 64)
        srow[i] = xc[(size_t)r * D_INNER + i];
    __syncthreads();

    const int NP = 2 * D_STATE + 1;          // 33
    for (int j = threadIdx.x; j < NP; j += 64) {
        float acc = 0.0f;
        for (int k = 0; k < D_INNER; ++k)
            acc = fmaf(srow[k], W_xproj[(size_t)k * NP + j], acc);
        if (j == NP - 1)                     // softplus(delta_raw)
            acc = (acc > 20.0f) ? acc : log1pf(__expf(acc));
        proj[(size_t)r * NP + j] = acc;
    }
}

// ---------------------------------------------------------------------------
// Selective scan + skip + gating.
// 16 state lanes per channel (2 channels per wave32); __shfl_xor tree reduces
// h*C over the 16-lane state group. B/C/delta/x/z chunks staged in LDS.
// Block = 256 threads = 16 channels of one batch; grid = BATCH * (D_INNER/16).
// ---------------------------------------------------------------------------
#define T_CHUNK 128
__global__ __launch_bounds__(256)
void scan_kernel(const float* __restrict__ xc, const float* __restrict__ proj,
                 const float* __restrict__ xz,   // z = xz[:, :, D_INNER:2*D_INNER]
                 const float* __restrict__ A_log, const float* __restrict__ D_param,
                 float* __restrict__ yg)
{
    const int NP = 2 * D_STATE + 1;          // 33
    __shared__ float sProj[T_CHUNK][NP];     // [t][0:16]=B, [16:32]=C, [32]=delta
    __shared__ float sX[T_CHUNK][16];
    __shared__ float sZ[T_CHUNK][16];

    const int grpPerB = D_INNER / 16;        // 128
    const int b    = blockIdx.x / grpPerB;
    const int dgrp = blockIdx.x % grpPerB;
    const int c    = threadIdx.x >> 4;       // channel within block: 0..15
    const int n    = threadIdx.x & 15;       // state index: 0..15
    const int d    = dgrp * 16 + c;

    const float A  = -__expf(A_log[(size_t)d * D_STATE + n]);
    const float Dp = D_param[d];
    float h = 0.0f;

    for (int t0 = 0; t0 < SEQLEN; t0 += T_CHUNK) {
        // Stage proj (B, C, delta) for this chunk
        for (int i = threadIdx.x; i < T_CHUNK * NP; i += 256) {
            int tt = i / NP, jj = i % NP;
            sProj[tt][jj] = proj[(size_t)(b * SEQLEN + t0 + tt) * NP + jj];
        }
        // Stage xc and z for the 16 channels
        for (int i = threadIdx.x; i < T_CHUNK * 16; i += 256) {
            int tt = i >> 4, cc = i & 15;
            size_t row = (size_t)(b * SEQLEN + t0 + tt);
            sX[tt][cc] = xc[row * D_INNER + dgrp * 16 + cc];
            sZ[tt][cc] = xz[row * (2 * D_INNER) + D_INNER + dgrp * 16 + cc];
        }
        __syncthreads();

        for (int tt = 0; tt < T_CHUNK; ++tt) {
            float delta = sProj[tt][2 * D_STATE];
            float x     = sX[tt][c];
            float dA    = __expf(delta * A);
            h = fmaf(dA, h, delta * sProj[tt][n] * x);      // h = dA*h + (d*B_n)*x
            float p = h * sProj[tt][D_STATE + n];           // h * C_n
            // reduce over the 16 state lanes (stays within 16-lane group)
            p += __shfl_xor(p, 8, 32);
            p += __shfl_xor(p, 4, 32);
            p += __shfl_xor(p, 2, 32);
            p += __shfl_xor(p, 1, 32);
            if (n == 0) {
                float y = p + x * Dp;                       // skip connection
                float z = sZ[tt][c];
                y *= z * sigmoid_f(z);                      // gate: y * silu(z)
                yg[(size_t)(b * SEQLEN + t0 + tt) * D_INNER + d] = y;
            }
        }
        __syncthreads();
    }
}

// ---------------------------------------------------------------------------
extern "C" void kernel_launch(void* const* d_in, const int* in_sizes, int n_in,
                              void* d_out, int out_size, void* d_ws, size_t ws_size,
                              hipStream_t stream)
{
    const float* x       = (const float*)d_in[0];
    const float* W_in    = (const float*)d_in[1];
    const float* b_in    = (const float*)d_in[2];
    const float* conv_w  = (const float*)d_in[3];
    const float* conv_b  = (const float*)d_in[4];
    const float* W_xproj = (const float*)d_in[5];
    const float* A_log   = (const float*)d_in[6];
    const float* D_param = (const float*)d_in[7];
    const float* W_out   = (const float*)d_in[8];
    const float* b_out   = (const float*)d_in[9];
    float* out = (float*)d_out;

    // Workspace layout (floats)
    float* xz   = (float*)d_ws;                          // NROWS x 2*D_INNER
    float* xc   = xz  + (size_t)NROWS * 2 * D_INNER;     // NROWS x D_INNER
    float* proj = xc  + (size_t)NROWS * D_INNER;         // NROWS x 33
    float* yg   = proj + (size_t)NROWS * (2 * D_STATE + 1); // NROWS x D_INNER

    // 1) xz = x @ W_in + b_in         (4096 x 1024) @ (1024 x 4096)
    {
        dim3 grid((2 * D_INNER) / 128, NROWS / 128);
        gemm_f32_wmma<128, 128, 16><<<grid, 256, 0, stream>>>(
            x, W_in, b_in, xz, NROWS, 2 * D_INNER, D_MODEL);
    }
    // 2) depthwise conv + SiLU
    {
        size_t total = (size_t)NROWS * D_INNER;
        conv_silu_kernel<<<(unsigned)((total + 255) / 256), 256, 0, stream>>>(
            xz, conv_w, conv_b, xc);
    }
    // 3) proj = xc @ W_xproj  (+ softplus on delta column)
    {
        xproj_kernel<<<NROWS, 64, 0, stream>>>(xc, W_xproj, proj);
    }
    // 4) selective scan + skip + gating
    {
        scan_kernel<<<BATCH * (D_INNER / 16), 256, 0, stream>>>(
            xc, proj, xz, A_log, D_param, yg);
    }
    // 5) out = yg @ W_out + b_out     (4096 x 2048) @ (2048 x 1024)
    {
        dim3 grid(D_MODEL / 128, NROWS / 128);
        gemm_f32_wmma<128, 128, 16><<<grid, 256, 0, stream>>>(
            yg, W_out, b_out, out, NROWS, D_MODEL, D_INNER);
    }
}